// GCNEncoder_73332271612105
// MI455X (gfx1250) — compile-verified
//
#include <hip/hip_runtime.h>

typedef float v2f __attribute__((ext_vector_type(2)));
typedef float v8f __attribute__((ext_vector_type(8)));

#define N_NODESC 100000
#define N_EDGESC 1600000
#define DIM 128

// ---------------- degree / norm ----------------

__global__ void k_init_deg(float* __restrict__ deg) {
  int i = blockIdx.x * blockDim.x + threadIdx.x;
  if (i < N_NODESC) deg[i] = 1.0f;  // self-loop contributes 1 to every node
}

__global__ void k_count_deg(const int* __restrict__ dst, float* __restrict__ deg) {
  int e = blockIdx.x * blockDim.x + threadIdx.x;
  if (e < N_EDGESC) unsafeAtomicAdd(&deg[dst[e]], 1.0f);
}

__global__ void k_dinv(const float* __restrict__ deg, float* __restrict__ dinv) {
  int i = blockIdx.x * blockDim.x + threadIdx.x;
  if (i < N_NODESC) dinv[i] = rsqrtf(deg[i]);  // deg >= 1 always
}

__global__ void k_zero4(float4* __restrict__ p, int n4) {
  int i = blockIdx.x * blockDim.x + threadIdx.x;
  if (i < n4) p[i] = make_float4(0.f, 0.f, 0.f, 0.f);
}

// ---------------- fp32 WMMA GEMM: H = X @ W  (N x 128) @ (128 x 128) ----------------
// One block = 16 output rows. 8 waves, wave w owns the 16x16 tile at columns [16w,16w+16).
// K swept in steps of 4 via V_WMMA_F32_16X16X4_F32.

__global__ __launch_bounds__(256) void k_gemm_wmma(const float* __restrict__ X,
                                                   const float* __restrict__ W,
                                                   float* __restrict__ H) {
  __shared__ float xs[16 * 132];  // 16x128 X tile, stride 132 -> conflict-free b64 reads

  const int t = threadIdx.x;
  const int wave = t >> 5;
  const int lane = t & 31;
  const int row0 = blockIdx.x * 16;  // 100000 = 6250*16, no remainder

  // Cooperative load of the 16x128 X tile: each thread moves 8 floats.
  {
    const int r = t >> 4;          // 0..15
    const int c = (t & 15) * 8;    // 0..120
    const float4* g = (const float4*)(X + (size_t)(row0 + r) * DIM + c);
    float4 p0 = g[0];
    float4 p1 = g[1];
    float4* s = (float4*)&xs[r * 132 + c];  // 132 % 4 == 0 -> 16B aligned
    s[0] = p0;
    s[1] = p1;
  }
  __syncthreads();

  const int m  = lane & 15;   // row within tile (A) / column within tile (B, C/D)
  const int hg = lane >> 4;   // half-wave group
  const int n0 = wave * 16;

  v8f acc = {};
  const float* wcol = W + n0 + m;  // column n0+m of W, row-major stride DIM

  #pragma unroll 4
  for (int kk = 0; kk < DIM; kk += 4) {
    // A fragment: lane l, vgpr v -> X[m][kk + v + 2*hg]  (ISA 16x4 f32 A layout)
    v2f a = *(const v2f*)&xs[m * 132 + kk + 2 * hg];
    // B fragment: lane l, vgpr v -> W[kk + v + 2*hg][n0 + m]
    v2f b;
    b.x = wcol[(size_t)(kk + 2 * hg + 0) * DIM];
    b.y = wcol[(size_t)(kk + 2 * hg + 1) * DIM];
    acc = __builtin_amdgcn_wmma_f32_16x16x4_f32(
        /*neg_a=*/false, a, /*neg_b=*/false, b,
        /*c_mod=*/(short)0, acc, /*reuse_a=*/false, /*reuse_b=*/false);
  }

  // C/D layout: vgpr v, lanes 0-15 -> M=v, lanes 16-31 -> M=v+8, N=lane&15.
  float* hrow = H + (size_t)row0 * DIM + n0 + m;
  #pragma unroll
  for (int v = 0; v < 8; ++v)
    hrow[(size_t)(v + 8 * hg) * DIM] = acc[v];
}

// ---------------- edge scatter: AGG[dst] += H[src] * dinv[src]*dinv[dst] ----------------
// One wave per edge, 4 features per lane, hardware f32 atomics into L2-resident AGG.

__global__ __launch_bounds__(256) void k_scatter(const int* __restrict__ src,
                                                 const int* __restrict__ dst,
                                                 const float* __restrict__ dinv,
                                                 const float* __restrict__ H,
                                                 float* __restrict__ AGG) {
  int e = blockIdx.x * 8 + (threadIdx.x >> 5);
  if (e >= N_EDGESC) return;
  int lane = threadIdx.x & 31;
  int s = src[e];
  int d = dst[e];
  float w = dinv[s] * dinv[d];
  float4 hv = ((const float4*)(H + (size_t)s * DIM))[lane];
  float* arow = AGG + (size_t)d * DIM + lane * 4;
  unsafeAtomicAdd(arow + 0, hv.x * w);
  unsafeAtomicAdd(arow + 1, hv.y * w);
  unsafeAtomicAdd(arow + 2, hv.z * w);
  unsafeAtomicAdd(arow + 3, hv.w * w);
}

// ---------------- finalize: OUT = [relu](AGG + H*dinv^2 + b) ----------------
// Self-loop term (weight dinv[i]^2) added here, non-atomically, after scatter completes.

__global__ void k_finalize(const float* __restrict__ AGG, const float* __restrict__ H,
                           const float* __restrict__ dinv, const float* __restrict__ b,
                           float* __restrict__ OUT, int do_relu) {
  int i = blockIdx.x * blockDim.x + threadIdx.x;
  if (i >= N_NODESC * DIM) return;
  int node = i >> 7;
  int f = i & (DIM - 1);
  float di = dinv[node];
  float v = AGG[i] + H[i] * di * di + b[f];
  if (do_relu) v = fmaxf(v, 0.0f);
  OUT[i] = v;
}

// ---------------- host launcher ----------------

extern "C" void kernel_launch(void* const* d_in, const int* in_sizes, int n_in,
                              void* d_out, int out_size, void* d_ws, size_t ws_size,
                              hipStream_t stream) {
  const float* x   = (const float*)d_in[0];
  const int*   ei  = (const int*)d_in[1];   // (2, E): row 0 = src, row 1 = dst
  const float* W1  = (const float*)d_in[2];
  const float* b1  = (const float*)d_in[3];
  const float* W2  = (const float*)d_in[4];
  const float* b2  = (const float*)d_in[5];
  float*       out = (float*)d_out;

  const int* src = ei;
  const int* dst = ei + N_EDGESC;

  // Workspace layout
  float* H    = (float*)d_ws;             // N*128
  float* AGG  = H + (size_t)N_NODESC * DIM;  // N*128
  float* deg  = AGG + (size_t)N_NODESC * DIM;
  float* dinv = deg + N_NODESC;

  const int feat_elems = N_NODESC * DIM;      // 12.8M
  const int feat_vec4  = feat_elems / 4;      // 3.2M

  dim3 blk(256);
  dim3 gNode((N_NODESC + 255) / 256);
  dim3 gEdge((N_EDGESC + 255) / 256);
  dim3 gGemm(N_NODESC / 16);                  // 6250
  dim3 gZero((feat_vec4 + 255) / 256);
  dim3 gScat(N_EDGESC / 8);                   // 8 edges (waves) per block
  dim3 gFin((feat_elems + 255) / 256);

  // Normalization
  k_init_deg<<<gNode, blk, 0, stream>>>(deg);
  k_count_deg<<<gEdge, blk, 0, stream>>>(dst, deg);
  k_dinv<<<gNode, blk, 0, stream>>>(deg, dinv);

  // ----- Layer 1 -----
  k_gemm_wmma<<<gGemm, blk, 0, stream>>>(x, W1, H);
  k_zero4<<<gZero, blk, 0, stream>>>((float4*)AGG, feat_vec4);
  k_scatter<<<gScat, blk, 0, stream>>>(src, dst, dinv, H, AGG);
  k_finalize<<<gFin, blk, 0, stream>>>(AGG, H, dinv, b1, out, /*relu=*/1);  // out = X2

  // ----- Layer 2 -----
  k_gemm_wmma<<<gGemm, blk, 0, stream>>>(out, W2, H);
  k_zero4<<<gZero, blk, 0, stream>>>((float4*)AGG, feat_vec4);
  k_scatter<<<gScat, blk, 0, stream>>>(src, dst, dinv, H, AGG);
  k_finalize<<<gFin, blk, 0, stream>>>(AGG, H, dinv, b2, out, /*relu=*/0);
}